// VectorQuantizer_36094905155774
// MI455X (gfx1250) — compile-verified
//
#include <hip/hip_runtime.h>
#include <hip/hip_bf16.h>

typedef __attribute__((ext_vector_type(2))) float v2f;
typedef __attribute__((ext_vector_type(4))) float v4f;
typedef __attribute__((ext_vector_type(8))) float v8f;

#define D_DIM 64
#define K_DIM 1024
#define N_TOTAL (128 * 32 * 32)          // 131072 latent vectors
#define OUT_ELEMS ((size_t)N_TOTAL * D_DIM)
#define ROWS_PER_BLOCK 128               // 8 waves x 16 rows
#define NUM_BLOCKS (N_TOTAL / ROWS_PER_BLOCK)  // 1024

// ---------------------------------------------------------------------------
// Kernel 1: codebook squared norms  eNorm[k] = sum_d E[d,k]^2
// ---------------------------------------------------------------------------
__global__ void vq_prep_kernel(const float* __restrict__ E,
                               float* __restrict__ eNorm) {
    int k = blockIdx.x * blockDim.x + threadIdx.x;
    if (k < K_DIM) {
        float s = 0.0f;
#pragma unroll
        for (int d = 0; d < D_DIM; ++d) {
            float e = E[d * K_DIM + k];
            s = fmaf(e, e, s);
        }
        eNorm[k] = s;
    }
}

// ---------------------------------------------------------------------------
// Kernel 2: main VQ kernel — fp32 WMMA GEMM + argmin + gather + loss partials
//   Wave owns a 16-row M tile. A fragment layout (V_WMMA_F32_16X16X4_F32):
//     lanes 0-15  hold row M=lane,    K = {4s+0, 4s+1}
//     lanes 16-31 hold row M=lane-16, K = {4s+2, 4s+3}
//   B fragment (4x16): lanes 0-15 col N=lane K={4s+0,4s+1};
//                      lanes 16-31 col N=lane-16 K={4s+2,4s+3}
//   C/D: VGPR j -> row M=j (lanes 0-15) / M=j+8 (lanes 16-31), N = lane&15
//   B fragments are double-buffered (b0/b1) so tile t+1's L2 loads overlap
//   tile t's serially-dependent 16-WMMA chain.
// ---------------------------------------------------------------------------
__global__ __launch_bounds__(256) void vq_main_kernel(
    const float* __restrict__ Z,      // [N_TOTAL, 64]
    const float* __restrict__ E,      // [64, 1024]
    const float* __restrict__ eNorm,  // [1024]
    float* __restrict__ Out,          // [N_TOTAL, 64]
    float* __restrict__ partial) {    // [NUM_BLOCKS]
    __shared__ float waveSums[8];

    const int tid  = threadIdx.x;
    const int wave = tid >> 5;
    const int lane = tid & 31;
    const int l15  = lane & 15;
    const int half = lane >> 4;                       // 0: rows j, 1: rows j+8
    const int waveRowBase = blockIdx.x * ROWS_PER_BLOCK + wave * 16;

    // ---- Preload A fragments for the wave's 16x64 tile (32 VGPRs) ----
    const float* zrow = Z + (size_t)(waveRowBase + l15) * D_DIM + 2 * half;
    v2f a[16];
#pragma unroll
    for (int s = 0; s < 16; ++s) {
        a[s] = *(const v2f*)(zrow + 4 * s);           // 8B-aligned global_load_b64
    }

    // ---- Running per-column argmin state ----
    float bestV[8];
    int   bestK[8];
#pragma unroll
    for (int j = 0; j < 8; ++j) { bestV[j] = 3.402823466e+38f; bestK[j] = 0; }

    // Per-lane base pointer into the codebook for B fragments.
    const float* bbase = E + (size_t)(2 * half) * K_DIM + l15;

    // Load the 16 B fragments of codebook tile t into dst[0..15].
    auto loadB = [&](int t, v2f* dst) {
        const float* bp = bbase + t * 16;
#pragma unroll
        for (int s = 0; s < 16; ++s) {
            dst[s].x = bp[(4 * s + 0) * K_DIM];
            dst[s].y = bp[(4 * s + 1) * K_DIM];
        }
    };

    // Run the 16-WMMA chain for tile t and fold scores into the running best.
    auto procTile = [&](int t, const v2f* bfrag) {
        v8f acc = {0.f, 0.f, 0.f, 0.f, 0.f, 0.f, 0.f, 0.f};
#pragma unroll
        for (int s = 0; s < 16; ++s) {
            acc = __builtin_amdgcn_wmma_f32_16x16x4_f32(
                /*neg_a=*/false, a[s], /*neg_b=*/false, bfrag[s],
                /*c_mod=*/(short)0, acc, /*reuse_a=*/false, /*reuse_b=*/false);
        }
        const int col = t * 16 + l15;
        const float en = eNorm[col];
#pragma unroll
        for (int j = 0; j < 8; ++j) {
            // score = ||e_col||^2 - 2 * <z_row, e_col>  (||z||^2 constant per row)
            float sc = fmaf(-2.0f, acc[j], en);
            if (sc < bestV[j]) { bestV[j] = sc; bestK[j] = col; }  // strict < keeps first k
        }
    };

    // ---- Sweep codebook: 64 tiles, 2x unrolled with double-buffered B ----
    v2f b0[16], b1[16];
    loadB(0, b0);
    for (int t = 0; t < 64; t += 2) {
        loadB(t + 1, b1);           // prefetch odd tile while even tile computes
        procTile(t, b0);
        loadB((t + 2) & 63, b0);    // prefetch next even tile (wraps harmlessly)
        procTile(t + 1, b1);
    }

    // ---- Cross-lane argmin within each 16-lane half (tie-break: smaller k) ----
#pragma unroll
    for (int j = 0; j < 8; ++j) {
#pragma unroll
        for (int m = 8; m >= 1; m >>= 1) {
            float ov = __shfl_xor(bestV[j], m, 32);
            int   ok = __shfl_xor(bestK[j], m, 32);
            if (ov < bestV[j] || (ov == bestV[j] && ok < bestK[j])) {
                bestV[j] = ov; bestK[j] = ok;
            }
        }
    }

    // ---- Gather winning codebook rows, write quantized output, loss partial ----
    float lsum = 0.0f;
#pragma unroll
    for (int j = 0; j < 8; ++j) {
        const int row   = waveRowBase + half * 8 + j;   // matches C/D VGPR layout
        const int enc   = bestK[j];                     // uniform within 16-lane half
        const int dbase = l15 * 4;                      // lane covers 4 channels
        v4f q4;
        q4.x = E[(size_t)(dbase + 0) * K_DIM + enc];
        q4.y = E[(size_t)(dbase + 1) * K_DIM + enc];
        q4.z = E[(size_t)(dbase + 2) * K_DIM + enc];
        q4.w = E[(size_t)(dbase + 3) * K_DIM + enc];
        const v4f zv = *(const v4f*)(Z + (size_t)row * D_DIM + dbase);
        // Output is write-once, never re-read: non-temporal store keeps the
        // codebook resident in WGP$/L2 instead of being evicted by the stream.
        __builtin_nontemporal_store(q4, (v4f*)(Out + (size_t)row * D_DIM + dbase));
        float dx = q4.x - zv.x, dy = q4.y - zv.y, dz = q4.z - zv.z, dw = q4.w - zv.w;
        lsum = fmaf(dx, dx, lsum);
        lsum = fmaf(dy, dy, lsum);
        lsum = fmaf(dz, dz, lsum);
        lsum = fmaf(dw, dw, lsum);
    }

    // ---- Deterministic block-level loss reduction (no float atomics) ----
#pragma unroll
    for (int m = 16; m >= 1; m >>= 1) lsum += __shfl_xor(lsum, m, 32);
    if (lane == 0) waveSums[wave] = lsum;
    __syncthreads();
    if (tid == 0) {
        float bs = 0.0f;
#pragma unroll
        for (int w = 0; w < 8; ++w) bs += waveSums[w];
        partial[blockIdx.x] = bs;
    }
}

// ---------------------------------------------------------------------------
// Kernel 3: fixed-order final reduction -> vq_loss = 1.25 * mean((q-z)^2)
// ---------------------------------------------------------------------------
__global__ void vq_finalize_kernel(const float* __restrict__ partial,
                                   float* __restrict__ lossOut) {
    if (blockIdx.x == 0 && threadIdx.x == 0) {
        float s = 0.0f;
        for (int i = 0; i < NUM_BLOCKS; ++i) s += partial[i];
        lossOut[0] = 1.25f * (s / (float)OUT_ELEMS);
    }
}

extern "C" void kernel_launch(void* const* d_in, const int* in_sizes, int n_in,
                              void* d_out, int out_size, void* d_ws, size_t ws_size,
                              hipStream_t stream) {
    const float* Z = (const float*)d_in[0];   // [128,32,32,64] = [131072,64]
    const float* E = (const float*)d_in[1];   // [64,1024]
    float* Out = (float*)d_out;               // quantized_st flat, then scalar loss

    float* eNorm   = (float*)d_ws;            // 1024 floats
    float* partial = eNorm + K_DIM;           // NUM_BLOCKS floats
    float* lossOut = Out + OUT_ELEMS;         // d_out[8388608]

    vq_prep_kernel<<<(K_DIM + 255) / 256, 256, 0, stream>>>(E, eNorm);
    vq_main_kernel<<<NUM_BLOCKS, 256, 0, stream>>>(Z, E, eNorm, Out, partial);
    vq_finalize_kernel<<<1, 32, 0, stream>>>(partial, lossOut);
}